// AttentionCell_70600672411879
// MI455X (gfx1250) — compile-verified
//
#include <hip/hip_runtime.h>

typedef __attribute__((ext_vector_type(2))) float v2f;
typedef __attribute__((ext_vector_type(8))) float v8f;

#define IMG_W 56
#define IMG_HW 3136          // 56*56
#define C_IN 64
#define BATCH 4
#define NPIX (BATCH * IMG_HW)   // 12544, multiple of 16; 3136 % 16 == 0 so tiles never cross batches

// ---------------------------------------------------------------------------
// Kernel A: grouped 1x1 conv for q/k/v via V_WMMA_F32_16X16X4_F32.
// One wave = one 16-pixel x 16-output-channel tile (two 8-ch groups packed
// block-diagonally), K=16 input channels = 4 WMMA k-steps of K=4.
// grid = (NPIX/16, 4 group-pairs, 3 matrices {q,k,v}), block = 32 (one wave).
// ---------------------------------------------------------------------------
__global__ __launch_bounds__(32)
void qkv_conv_wmma(const float* __restrict__ x,
                   const float* __restrict__ wq,
                   const float* __restrict__ wk,
                   const float* __restrict__ bk,
                   const float* __restrict__ wv,
                   const float* __restrict__ bv,
                   float* __restrict__ qws,
                   float* __restrict__ kws,
                   float* __restrict__ vws)
{
    const int lane  = threadIdx.x;        // 0..31
    const int ptile = blockIdx.x;         // 0..783
    const int gpair = blockIdx.y;         // 0..3 -> output channels gpair*16 .. +15
    const int mat   = blockIdx.z;         // 0=q, 1=k, 2=v

    const float* w    = (mat == 0) ? wq : (mat == 1) ? wk : wv;   // (G=8, O=8, I=8)
    const float* bias = (mat == 1) ? bk : (mat == 2) ? bv : nullptr;
    float* outp       = (mat == 0) ? qws : (mat == 1) ? kws : vws;

    const int p0   = ptile * 16;
    const int b    = p0 / IMG_HW;         // tile stays inside one batch image
    const int pix0 = p0 - b * IMG_HW;

    const int mrow = lane & 15;           // pixel row of A / column of B
    const int half = lane >> 4;           // 0: lanes 0-15 (K 0,1), 1: lanes 16-31 (K 2,3)

    // B[i16][n] is block-diagonal: group gpair*2 in the top-left 8x8,
    // group gpair*2+1 in the bottom-right 8x8.  Preload all 16 K-rows.
    const int n = mrow;                   // output channel within tile
    float breg[8];
#pragma unroll
    for (int kk = 0; kk < 4; ++kk) {
#pragma unroll
        for (int t = 0; t < 2; ++t) {
            const int i16 = kk * 4 + half * 2 + t;  // input channel within tile
            const int gi = i16 >> 3, go = n >> 3;
            const int g = gpair * 2 + gi;
            const float wval = w[g * 64 + (n & 7) * 8 + (i16 & 7)];
            breg[kk * 2 + t] = (gi == go) ? wval : 0.0f;
        }
    }

    v8f c = {};
    const float* xb = x + (size_t)b * C_IN * IMG_HW + pix0 + mrow;
#pragma unroll
    for (int kk = 0; kk < 4; ++kk) {
        const int ci = gpair * 16 + kk * 4 + half * 2;  // absolute input channel
        v2f a;
        a.x = xb[(size_t)ci * IMG_HW];
        a.y = xb[(size_t)(ci + 1) * IMG_HW];
        v2f bm;
        bm.x = breg[kk * 2 + 0];
        bm.y = breg[kk * 2 + 1];
        // D = A(16x4) x B(4x16) + C ; 8 args: neg_a, A, neg_b, B, c_mod, C, reuse_a, reuse_b
        c = __builtin_amdgcn_wmma_f32_16x16x4_f32(false, a, false, bm,
                                                  (short)0, c, false, false);
    }

    // D layout: VGPR j = row (j + half*8), column n (lanes 0-15 / 16-31 mirror).
    const int cout = gpair * 16 + n;
    const float badd = bias ? bias[cout] : 0.0f;
    float* op = outp + (size_t)(b * C_IN + cout) * IMG_HW + pix0;
#pragma unroll
    for (int j = 0; j < 8; ++j) {
        const int m = j + half * 8;
        op[m] = c[j] + badd;
    }
}

// ---------------------------------------------------------------------------
// Kernel B: per-channel local attention.  logits = q*(k_win + r),
// r = rel_x[d][ki] (d<4) or rel_y[d-4][kj] (d>=4).  Out-of-image window
// positions use the conv bias (conv of zero padding + bias).
// One thread per (b, c, pixel).
// ---------------------------------------------------------------------------
__global__ __launch_bounds__(256)
void local_attn_softmax(const float* __restrict__ qws,
                        const float* __restrict__ kws,
                        const float* __restrict__ vws,
                        const float* __restrict__ bk,
                        const float* __restrict__ bv,
                        const float* __restrict__ relx,  // (4,7,1) flat: d*7+ki
                        const float* __restrict__ rely,  // (4,1,7) flat: d*7+kj
                        float* __restrict__ out)
{
    const int idx = blockIdx.x * blockDim.x + threadIdx.x;
    if (idx >= BATCH * C_IN * IMG_HW) return;

    const int pix = idx % IMG_HW;
    const int c   = (idx / IMG_HW) % C_IN;
    const int b   = idx / (C_IN * IMG_HW);
    const int y   = pix / IMG_W;
    const int xc  = pix % IMG_W;
    const int d   = c & 7;
    const bool useRow = (d < 4);          // d<4 -> rel_x over ki ; d>=4 -> rel_y over kj

    float rv[7];
#pragma unroll
    for (int i = 0; i < 7; ++i)
        rv[i] = useRow ? relx[d * 7 + i] : rely[(d - 4) * 7 + i];

    const float q   = qws[idx];
    const float bkc = bk[c];
    const float bvc = bv[c];
    const float* kc = kws + (size_t)(b * C_IN + c) * IMG_HW;
    const float* vc = vws + (size_t)(b * C_IN + c) * IMG_HW;

    float logits[49];
    float mx = -3.402823466e38f;
#pragma unroll
    for (int ki = 0; ki < 7; ++ki) {
        const int yy = y + ki - 3;
#pragma unroll
        for (int kj = 0; kj < 7; ++kj) {
            const int xx = xc + kj - 3;
            const bool inb = (yy >= 0) & (yy < IMG_W) & (xx >= 0) & (xx < IMG_W);
            const float kval = inb ? kc[yy * IMG_W + xx] : bkc;
            const float rr = useRow ? rv[ki] : rv[kj];
            const float l = q * (kval + rr);
            logits[ki * 7 + kj] = l;
            mx = fmaxf(mx, l);
        }
    }

    float s = 0.0f, acc = 0.0f;
#pragma unroll
    for (int ki = 0; ki < 7; ++ki) {
        const int yy = y + ki - 3;
#pragma unroll
        for (int kj = 0; kj < 7; ++kj) {
            const int xx = xc + kj - 3;
            const bool inb = (yy >= 0) & (yy < IMG_W) & (xx >= 0) & (xx < IMG_W);
            const float vval = inb ? vc[yy * IMG_W + xx] : bvc;
            const float e = __expf(logits[ki * 7 + kj] - mx);
            s   += e;
            acc += e * vval;
        }
    }

    out[idx] = acc / s;
}

// ---------------------------------------------------------------------------
extern "C" void kernel_launch(void* const* d_in, const int* in_sizes, int n_in,
                              void* d_out, int out_size, void* d_ws, size_t ws_size,
                              hipStream_t stream)
{
    const float* x    = (const float*)d_in[0];   // (4,64,56,56)
    const float* wq   = (const float*)d_in[1];   // (8,8,8)
    const float* wk   = (const float*)d_in[2];   // (8,8,8)
    const float* bk   = (const float*)d_in[3];   // (64)
    const float* wv   = (const float*)d_in[4];   // (8,8,8)
    const float* bv   = (const float*)d_in[5];   // (64)
    const float* relx = (const float*)d_in[6];   // (4,7,1)
    const float* rely = (const float*)d_in[7];   // (4,1,7)
    float* out = (float*)d_out;

    const size_t plane = (size_t)BATCH * C_IN * IMG_HW;  // 802816 floats
    float* qws = (float*)d_ws;
    float* kws = qws + plane;
    float* vws = kws + plane;

    // q/k/v grouped 1x1 convs via WMMA
    dim3 gridA(NPIX / 16, 4, 3);   // 784 pixel-tiles x 4 group-pairs x {q,k,v}
    qkv_conv_wmma<<<gridA, 32, 0, stream>>>(x, wq, wk, bk, wv, bv, qws, kws, vws);

    // per-channel 7x7 softmax attention
    const int total = BATCH * C_IN * IMG_HW;
    local_attn_softmax<<<(total + 255) / 256, 256, 0, stream>>>(
        qws, kws, vws, bk, bv, relx, rely, out);
}